// GATNodeBaseline_1065151889635
// MI455X (gfx1250) — compile-verified
//
#include <hip/hip_runtime.h>
#include <math.h>

#define DFEAT 64
#define NEG_SLOPE 0.2f

typedef __attribute__((ext_vector_type(2))) float v2f;
typedef __attribute__((ext_vector_type(8))) float v8f;

// ---------------------------------------------------------------------------
// GEMM: C[M,Ncol] = act(A[M,K] @ B[K,Ncol] + bias)   (f32 WMMA 16x16x4)
// One wave computes a 16(M) x 64(N) strip: 4 accumulators, 4 WMMAs per K-step.
// A frag (16x4 f32): lane m = lane&15, reg r -> K = 4*kt + 2*(lane>>4) + r
// B frag (4x16 f32): lane n = lane&15, reg r -> K = 4*kt + 2*(lane>>4) + r
// C frag (16x16 f32): lane n = lane&15, reg r -> M = r + 8*(lane>>4)
// Main K-loop is fully unguarded (tail peeled) so no EXEC manipulation occurs
// between WMMAs.
// ---------------------------------------------------------------------------
__global__ __launch_bounds__(128)
void gemm_f32_wmma(const float* __restrict__ A, const float* __restrict__ B,
                   const float* __restrict__ bias, float* __restrict__ C,
                   int M, int K, int Ncol, int do_relu, int has_bias)
{
    const int lane = threadIdx.x & 31;
    const int wave = threadIdx.x >> 5;
    const int mt   = blockIdx.x * 4 + wave;
    if (mt * 16 >= M) return;                 // whole-wave uniform exit
    const int hl    = lane >> 4;              // 0 or 1
    const int ml    = lane & 15;
    const int nbase = blockIdx.y * 64;

    v8f acc[4];
    acc[0] = (v8f){}; acc[1] = (v8f){}; acc[2] = (v8f){}; acc[3] = (v8f){};

    int row = mt * 16 + ml;
    if (row >= M) row = M - 1;                // clamp (results discarded on store)

    const float* __restrict__ ap = A + (size_t)row * K + 2 * hl;
    const float* __restrict__ bp = B + (size_t)(2 * hl) * Ncol + nbase + ml;
    const size_t bstep = (size_t)4 * Ncol;

    const int kFull = K >> 2;                 // unguarded full tiles
    for (int kt = 0; kt < kFull; ++kt) {
        v2f a;
        a.x = ap[0];
        a.y = ap[1];
#pragma unroll
        for (int j = 0; j < 4; ++j) {
            v2f b;
            b.x = bp[j * 16];
            b.y = bp[(size_t)Ncol + j * 16];
            acc[j] = __builtin_amdgcn_wmma_f32_16x16x4_f32(
                false, a, false, b, (short)0, acc[j], false, false);
        }
        ap += 4;
        bp += bstep;
    }
    if (K & 3) {                              // single guarded tail tile
        const int kb = (kFull << 2) + 2 * hl;
        v2f a;
        a.x = (kb     < K) ? ap[0] : 0.f;
        a.y = (kb + 1 < K) ? ap[1] : 0.f;
#pragma unroll
        for (int j = 0; j < 4; ++j) {
            v2f b;
            b.x = (kb     < K) ? bp[j * 16] : 0.f;
            b.y = (kb + 1 < K) ? bp[(size_t)Ncol + j * 16] : 0.f;
            acc[j] = __builtin_amdgcn_wmma_f32_16x16x4_f32(
                false, a, false, b, (short)0, acc[j], false, false);
        }
    }

#pragma unroll
    for (int j = 0; j < 4; ++j) {
        const int n  = nbase + j * 16 + ml;
        const float bv = has_bias ? bias[n] : 0.f;
#pragma unroll
        for (int r = 0; r < 8; ++r) {
            const int m = mt * 16 + r + 8 * hl;
            if (m < M) {
                float v = acc[j][r] + bv;
                if (do_relu) v = fmaxf(v, 0.f);
                C[(size_t)m * Ncol + n] = v;
            }
        }
    }
}

// ---------------------------------------------------------------------------
// a_s[n,h] = <h[n,h,:], att_src[h,:]> ; a_d likewise
// ---------------------------------------------------------------------------
__global__ void att_scores(const float* __restrict__ hfeat,
                           const float* __restrict__ att_s,
                           const float* __restrict__ att_d,
                           float* __restrict__ as, float* __restrict__ ad,
                           int N, int H)
{
    int t = blockIdx.x * blockDim.x + threadIdx.x;
    if (t >= N * H) return;
    const int n = t / H, hh = t % H;
    const float* hp = hfeat + (size_t)n * H * DFEAT + hh * DFEAT;
    const float* sp = att_s + hh * DFEAT;
    const float* dp = att_d + hh * DFEAT;
    float ss = 0.f, sd = 0.f;
#pragma unroll 8
    for (int d = 0; d < DFEAT; ++d) { float v = hp[d]; ss += v * sp[d]; sd += v * dp[d]; }
    as[t] = ss; ad[t] = sd;
}

__global__ void fill_f32(float* __restrict__ p, float v, int n)
{
    int t = blockIdx.x * blockDim.x + threadIdx.x;
    if (t < n) p[t] = v;
}

__device__ __forceinline__ void atomicMaxF(float* addr, float value)
{
    if (value >= 0.f)
        atomicMax((int*)addr, __float_as_int(value));
    else
        atomicMin((unsigned int*)addr, __float_as_uint(value));
}

__device__ __forceinline__ float leaky(float e)
{
    return e > 0.f ? e : NEG_SLOPE * e;
}

// Pass A: segment max of leaky_relu(a_s[src]+a_d[dst]) per dst,head
__global__ void edge_max(const long long* __restrict__ ei, int E, int N,
                         const float* __restrict__ as, const float* __restrict__ ad,
                         float* __restrict__ emax, int H)
{
    int idx = blockIdx.x * blockDim.x + threadIdx.x;
    if (idx >= E + N) return;
    int s, d;
    if (idx < E) { s = (int)ei[idx]; d = (int)ei[E + idx]; } else { s = d = idx - E; }
    for (int h = 0; h < H; ++h) {
        float e = leaky(as[s * H + h] + ad[d * H + h]);
        atomicMaxF(&emax[d * H + h], e);
    }
}

// Pass B: denom[dst,h] += exp(e - emax[dst,h])
__global__ void edge_expsum(const long long* __restrict__ ei, int E, int N,
                            const float* __restrict__ as, const float* __restrict__ ad,
                            const float* __restrict__ emax, float* __restrict__ denom, int H)
{
    int idx = blockIdx.x * blockDim.x + threadIdx.x;
    if (idx >= E + N) return;
    int s, d;
    if (idx < E) { s = (int)ei[idx]; d = (int)ei[E + idx]; } else { s = d = idx - E; }
    for (int h = 0; h < H; ++h) {
        float e = leaky(as[s * H + h] + ad[d * H + h]);
        atomicAdd(&denom[d * H + h], __expf(e - emax[d * H + h]));
    }
}

// Pass C: out[dst,h,:] += alpha * h[src,h,:]   (one wave per edge)
__global__ void edge_aggregate(const long long* __restrict__ ei, int E, int N,
                               const float* __restrict__ as, const float* __restrict__ ad,
                               const float* __restrict__ emax, const float* __restrict__ denom,
                               const float* __restrict__ hfeat, float* __restrict__ out, int H)
{
    const int lane = threadIdx.x & 31;
    const int widx = blockIdx.x * (blockDim.x >> 5) + (threadIdx.x >> 5);
    if (widx >= E + N) return;
    int s, d;
    if (widx < E) { s = (int)ei[widx]; d = (int)ei[E + widx]; } else { s = d = widx - E; }

    float alpha_l = 0.f;
    if (lane < H) {
        float e = leaky(as[s * H + lane] + ad[d * H + lane]);
        alpha_l = __expf(e - emax[d * H + lane]) / denom[d * H + lane];
    }
    const int F = H * DFEAT;
    const float* __restrict__ src = hfeat + (size_t)s * F;
    float* __restrict__ dst = out + (size_t)d * F;
    for (int i = lane; i < F; i += 32) {
        const float alpha = __shfl(alpha_l, i >> 6, 32);   // head = i / 64
        atomicAdd(&dst[i], src[i] * alpha);
    }
}

__global__ void bias_relu(float* __restrict__ x, const float* __restrict__ b, int total, int F)
{
    int t = blockIdx.x * blockDim.x + threadIdx.x;
    if (t >= total) return;
    x[t] = fmaxf(x[t] + b[t % F], 0.f);
}

__global__ void classify(const float* __restrict__ h2, const float* __restrict__ clsW,
                         const float* __restrict__ clsb, float* __restrict__ out, int N)
{
    int n = blockIdx.x * blockDim.x + threadIdx.x;
    if (n >= N) return;
    const float* hp = h2 + (size_t)n * DFEAT;
    float s = clsb[0];
#pragma unroll 8
    for (int d = 0; d < DFEAT; ++d) s += hp[d] * clsW[d];
    out[n] = s;
}

// ---------------------------------------------------------------------------
extern "C" void kernel_launch(void* const* d_in, const int* in_sizes, int n_in,
                              void* d_out, int out_size, void* d_ws, size_t ws_size,
                              hipStream_t stream)
{
    const float*      x      = (const float*)d_in[0];
    const long long*  ei     = (const long long*)d_in[1];
    const float*      projW  = (const float*)d_in[2];
    const float*      projb  = (const float*)d_in[3];
    const float*      W1     = (const float*)d_in[4];
    const float*      attS1  = (const float*)d_in[5];
    const float*      attD1  = (const float*)d_in[6];
    const float*      b1     = (const float*)d_in[7];
    const float*      W2     = (const float*)d_in[8];
    const float*      attS2  = (const float*)d_in[9];
    const float*      attD2  = (const float*)d_in[10];
    const float*      b2     = (const float*)d_in[11];
    const float*      clsW   = (const float*)d_in[12];
    const float*      clsb   = (const float*)d_in[13];
    float*            outp   = (float*)d_out;

    const int Fin = in_sizes[2] / DFEAT;       // 165
    const int N   = in_sizes[0] / Fin;         // 50000
    const int E   = in_sizes[1] / 2;           // 800000
    const int H1  = in_sizes[5] / DFEAT;       // 4
    const int F1  = H1 * DFEAT;                // 256

    // ---- workspace bump allocator (256B aligned) ----
    char* w = (char*)d_ws;
    size_t off = 0;
    auto take = [&](size_t bytes) -> float* {
        float* p = (float*)(w + off);
        off += (bytes + 255) & ~(size_t)255;
        return p;
    };
    float* h0    = take((size_t)N * DFEAT * 4);   // proj output       (reused as hW2)
    float* h1    = take((size_t)N * F1 * 4);      // GAT1 linear       (reused as out2)
    float* as1   = take((size_t)N * H1 * 4);
    float* ad1   = take((size_t)N * H1 * 4);
    float* emax1 = take((size_t)N * H1 * 4);
    float* den1  = take((size_t)N * H1 * 4);
    float* out1  = take((size_t)N * F1 * 4);
    float* as2   = take((size_t)N * 4);
    float* ad2   = take((size_t)N * 4);
    float* emax2 = take((size_t)N * 4);
    float* den2  = take((size_t)N * 4);
    float* hW2   = h0;   // lifetime-disjoint reuse
    float* out2  = h1;   // lifetime-disjoint reuse
    (void)ws_size; (void)n_in; (void)out_size;

    const int Mtiles = (N + 15) / 16;
    const dim3 gblk(128);
    const int TOT = E + N;

    // 1) h0 = relu(x @ projW + projb)           [N,165]x[165,64]
    gemm_f32_wmma<<<dim3((Mtiles + 3) / 4, 1), gblk, 0, stream>>>(
        x, projW, projb, h0, N, Fin, DFEAT, 1, 1);

    // 2) h1 = h0 @ W1                           [N,64]x[64,256]
    gemm_f32_wmma<<<dim3((Mtiles + 3) / 4, F1 / 64), gblk, 0, stream>>>(
        h0, W1, nullptr, h1, N, DFEAT, F1, 0, 0);

    // 3) GAT layer 1 edge phase
    att_scores<<<(N * H1 + 255) / 256, 256, 0, stream>>>(h1, attS1, attD1, as1, ad1, N, H1);
    fill_f32<<<(N * H1 + 255) / 256, 256, 0, stream>>>(emax1, -INFINITY, N * H1);
    fill_f32<<<(N * H1 + 255) / 256, 256, 0, stream>>>(den1, 0.f, N * H1);
    fill_f32<<<(N * F1 + 255) / 256, 256, 0, stream>>>(out1, 0.f, N * F1);
    edge_max<<<(TOT + 255) / 256, 256, 0, stream>>>(ei, E, N, as1, ad1, emax1, H1);
    edge_expsum<<<(TOT + 255) / 256, 256, 0, stream>>>(ei, E, N, as1, ad1, emax1, den1, H1);
    edge_aggregate<<<(TOT + 7) / 8, 256, 0, stream>>>(ei, E, N, as1, ad1, emax1, den1, h1, out1, H1);
    bias_relu<<<(N * F1 + 255) / 256, 256, 0, stream>>>(out1, b1, N * F1, F1);

    // 4) hW2 = out1 @ W2                        [N,256]x[256,64]
    gemm_f32_wmma<<<dim3((Mtiles + 3) / 4, 1), gblk, 0, stream>>>(
        out1, W2, nullptr, hW2, N, F1, DFEAT, 0, 0);

    // 5) GAT layer 2 edge phase (H=1)
    att_scores<<<(N + 255) / 256, 256, 0, stream>>>(hW2, attS2, attD2, as2, ad2, N, 1);
    fill_f32<<<(N + 255) / 256, 256, 0, stream>>>(emax2, -INFINITY, N);
    fill_f32<<<(N + 255) / 256, 256, 0, stream>>>(den2, 0.f, N);
    fill_f32<<<(N * DFEAT + 255) / 256, 256, 0, stream>>>(out2, 0.f, N * DFEAT);
    edge_max<<<(TOT + 255) / 256, 256, 0, stream>>>(ei, E, N, as2, ad2, emax2, 1);
    edge_expsum<<<(TOT + 255) / 256, 256, 0, stream>>>(ei, E, N, as2, ad2, emax2, den2, 1);
    edge_aggregate<<<(TOT + 7) / 8, 256, 0, stream>>>(ei, E, N, as2, ad2, emax2, den2, hW2, out2, 1);
    bias_relu<<<(N * DFEAT + 255) / 256, 256, 0, stream>>>(out2, b2, N * DFEAT, DFEAT);

    // 6) classifier
    classify<<<(N + 255) / 256, 256, 0, stream>>>(out2, clsW, clsb, outp, N);
}